// UpsampleLoss_88957362635530
// MI455X (gfx1250) — compile-verified
//
#include <hip/hip_runtime.h>
#include <math.h>

// ---------------------------------------------------------------------------
// UpsampleLoss on MI455X (gfx1250, wave32).
// Distance tiles generated with V_WMMA_F32_16X16X4_F32 via the norm trick:
//   d2 = A(k=[ax,ay,az,|a|^2]) . B(k=[-2bx,-2by,-2bz,1]) + C(=|b|^2 per column)
// Cross-lane reductions use DPP16 xor-butterflies (quad_perm / row mirrors)
// instead of ds_bpermute, keeping the epilogues entirely in the VALU pipe.
// ---------------------------------------------------------------------------

typedef __attribute__((ext_vector_type(2))) float v2f;
typedef __attribute__((ext_vector_type(8))) float v8f;

#define BATCH    8
#define NPTS     4096
#define TILE     16
#define WAVES    8               // 256-thread block = 8 wave32s
#define COLTILES (NPTS / TILE)   // 256

// DPP controls for xor-butterflies within a 16-lane row.
#define DPP_XOR1 0xB1   // QUAD_PERM([1,0,3,2])
#define DPP_XOR2 0x4E   // QUAD_PERM([2,3,0,1])
#define DPP_XOR4 0x141  // ROW_HALF_MIRROR (== xor4 once quads are uniform)
#define DPP_XOR8 0x140  // ROW_MIRROR      (== xor8 once octs are uniform)

template <int CTRL>
__device__ __forceinline__ float dpp_bfly(float x) {
  const int xi = __float_as_int(x);
  const int r  = __builtin_amdgcn_update_dpp(xi, xi, CTRL, 0xF, 0xF, true);
  return __int_as_float(r);
}

__device__ __forceinline__ v8f wmma_d2_tile(v2f a, v2f b, v8f c) {
  // 8 args: (neg_a, A, neg_b, B, c_mod, C, reuse_a, reuse_b)
  return __builtin_amdgcn_wmma_f32_16x16x4_f32(
      false, a, false, b, (short)0, c, false, false);
}

// Branchless insert keeping the 5 smallest, sorted ascending.
__device__ __forceinline__ void insert5(float (&t)[5], float v) {
  float lo, c;
  lo = fminf(v, t[0]); c = fmaxf(v, t[0]); t[0] = lo;
  lo = fminf(c, t[1]); c = fmaxf(c, t[1]); t[1] = lo;
  lo = fminf(c, t[2]); c = fmaxf(c, t[2]); t[2] = lo;
  lo = fminf(c, t[3]); c = fmaxf(c, t[3]); t[3] = lo;
  t[4] = fminf(c, t[4]);
}

// One xor-butterfly merge step of the per-lane sorted 5-lists (all 8 row slots).
template <int CTRL>
__device__ __forceinline__ void merge_step(float (&t)[8][5]) {
#pragma unroll
  for (int i = 0; i < 8; ++i) {
    const float o0 = dpp_bfly<CTRL>(t[i][0]);
    const float o1 = dpp_bfly<CTRL>(t[i][1]);
    const float o2 = dpp_bfly<CTRL>(t[i][2]);
    const float o3 = dpp_bfly<CTRL>(t[i][3]);
    const float o4 = dpp_bfly<CTRL>(t[i][4]);
    insert5(t[i], o0); insert5(t[i], o1); insert5(t[i], o2);
    insert5(t[i], o3); insert5(t[i], o4);
  }
}

// ---------------------------------------------------------------------------
// One directed chamfer pass: rows = A, cols = Bm. Each block handles 16 rows
// of one batch; accumulates  weight * sum_rows(min_col d2) / radius[b].
// ---------------------------------------------------------------------------
__global__ __launch_bounds__(256) void chamfer_min_kernel(
    const float* __restrict__ A, const float* __restrict__ Bm,
    const float* __restrict__ radius, float* __restrict__ acc, float weight)
{
  const int b       = blockIdx.x / COLTILES;
  const int rowBase = (blockIdx.x % COLTILES) * TILE;
  const int lane    = threadIdx.x & 31;
  const int w       = threadIdx.x >> 5;
  const int lcol    = lane & 15;
  const bool hi     = lane >= 16;

  const float* Ab = A  + (size_t)b * NPTS * 3;
  const float* Bb = Bm + (size_t)b * NPTS * 3;

  // A fragment (persistent across the column sweep)
  const int row = rowBase + lcol;
  const float ax = Ab[row * 3 + 0], ay = Ab[row * 3 + 1], az = Ab[row * 3 + 2];
  const float na = fmaf(ax, ax, fmaf(ay, ay, az * az));
  v2f afrag;
  afrag.x = hi ? az : ax;   // K2 : K0
  afrag.y = hi ? na : ay;   // K3 : K1

  float rowmin[8];
#pragma unroll
  for (int i = 0; i < 8; ++i) rowmin[i] = __builtin_inff();

  for (int ct = w; ct < COLTILES; ct += WAVES) {
    const int col = ct * TILE + lcol;
    const float bx = Bb[col * 3 + 0], by = Bb[col * 3 + 1], bz = Bb[col * 3 + 2];
    const float nb = fmaf(bx, bx, fmaf(by, by, bz * bz));
    v2f bfrag;
    bfrag.x = hi ? (-2.0f * bz) : (-2.0f * bx);  // K2 : K0
    bfrag.y = hi ? 1.0f         : (-2.0f * by);  // K3 : K1
    v8f c;
#pragma unroll
    for (int i = 0; i < 8; ++i) c[i] = nb;       // |b|^2 broadcast per column
    v8f d = wmma_d2_tile(afrag, bfrag, c);
#pragma unroll
    for (int i = 0; i < 8; ++i) rowmin[i] = fminf(rowmin[i], d[i]);
  }

  // Min across the 16 column lanes of each half-wave (pure DPP, no LDS).
#pragma unroll
  for (int i = 0; i < 8; ++i) {
    rowmin[i] = fminf(rowmin[i], dpp_bfly<DPP_XOR1>(rowmin[i]));
    rowmin[i] = fminf(rowmin[i], dpp_bfly<DPP_XOR2>(rowmin[i]));
    rowmin[i] = fminf(rowmin[i], dpp_bfly<DPP_XOR4>(rowmin[i]));
    rowmin[i] = fminf(rowmin[i], dpp_bfly<DPP_XOR8>(rowmin[i]));
  }

  __shared__ float red[16][WAVES];
  if (lcol == 0) {
    const int rbase = hi ? 8 : 0;
#pragma unroll
    for (int i = 0; i < 8; ++i) red[rbase + i][w] = rowmin[i];
  }
  __syncthreads();

  __shared__ float ssum[16];
  if (threadIdx.x < 16) {
    float m = red[threadIdx.x][0];
#pragma unroll
    for (int ww = 1; ww < WAVES; ++ww) m = fminf(m, red[threadIdx.x][ww]);
    ssum[threadIdx.x] = m;
  }
  __syncthreads();
  if (threadIdx.x == 0) {
    const float invr = 1.0f / radius[b];
    float s = 0.0f;
#pragma unroll
    for (int i = 0; i < 16; ++i) s += ssum[i];
    atomicAdd(&acc[0], weight * invr * s);
  }
}

// ---------------------------------------------------------------------------
// Uniformity: per pred point, 4 nearest non-self distances (top-5 incl self).
// ---------------------------------------------------------------------------
__global__ __launch_bounds__(256) void uniform_topk_kernel(
    const float* __restrict__ P, float* __restrict__ acc)
{
  const int b       = blockIdx.x / COLTILES;
  const int rowBase = (blockIdx.x % COLTILES) * TILE;
  const int lane    = threadIdx.x & 31;
  const int w       = threadIdx.x >> 5;
  const int lcol    = lane & 15;
  const bool hi     = lane >= 16;

  const float* Pb = P + (size_t)b * NPTS * 3;

  const int row = rowBase + lcol;
  const float ax = Pb[row * 3 + 0], ay = Pb[row * 3 + 1], az = Pb[row * 3 + 2];
  const float na = fmaf(ax, ax, fmaf(ay, ay, az * az));
  v2f afrag;
  afrag.x = hi ? az : ax;
  afrag.y = hi ? na : ay;

  float top[8][5];
#pragma unroll
  for (int i = 0; i < 8; ++i)
#pragma unroll
    for (int j = 0; j < 5; ++j) top[i][j] = __builtin_inff();

  for (int ct = w; ct < COLTILES; ct += WAVES) {
    const int col = ct * TILE + lcol;
    const float bx = Pb[col * 3 + 0], by = Pb[col * 3 + 1], bz = Pb[col * 3 + 2];
    const float nb = fmaf(bx, bx, fmaf(by, by, bz * bz));
    v2f bfrag;
    bfrag.x = hi ? (-2.0f * bz) : (-2.0f * bx);
    bfrag.y = hi ? 1.0f         : (-2.0f * by);
    v8f c;
#pragma unroll
    for (int i = 0; i < 8; ++i) c[i] = nb;
    v8f d = wmma_d2_tile(afrag, bfrag, c);
#pragma unroll
    for (int i = 0; i < 8; ++i) insert5(top[i], d[i]);
  }

  // Symmetric xor-butterfly list-merge across the 16 column lanes (DPP only).
  merge_step<DPP_XOR1>(top);
  merge_step<DPP_XOR2>(top);
  merge_step<DPP_XOR4>(top);
  merge_step<DPP_XOR8>(top);

  __shared__ float wtop[16][WAVES][5];
  if (lcol == 0) {
    const int rbase = hi ? 8 : 0;
#pragma unroll
    for (int i = 0; i < 8; ++i)
#pragma unroll
      for (int j = 0; j < 5; ++j) wtop[rbase + i][w][j] = top[i][j];
  }
  __syncthreads();

  __shared__ float ssum[16];
  if (threadIdx.x < 16) {
    float t5[5];
#pragma unroll
    for (int j = 0; j < 5; ++j) t5[j] = __builtin_inff();
    for (int ww = 0; ww < WAVES; ++ww)
#pragma unroll
      for (int j = 0; j < 5; ++j) insert5(t5, wtop[threadIdx.x][ww][j]);

    // t5[0] ~ 0 is the self distance; use neighbors 1..4.
    const float inv_h2 = 1.0f / (0.03f * 0.03f);
    float s = 0.0f;
#pragma unroll
    for (int k = 1; k < 5; ++k) {
      const float d2   = fmaxf(t5[k], 1e-12f);
      const float dist = sqrtf(d2);
      s += (0.07f - dist) * __expf(-d2 * inv_h2);
    }
    ssum[threadIdx.x] = s;
  }
  __syncthreads();
  if (threadIdx.x == 0) {
    float s = 0.0f;
#pragma unroll
    for (int i = 0; i < 16; ++i) s += ssum[i];
    atomicAdd(&acc[1], s);
  }
}

__global__ void init_kernel(float* __restrict__ acc) {
  if (threadIdx.x < 2) acc[threadIdx.x] = 0.0f;
}

__global__ void finalize_kernel(const float* __restrict__ acc,
                                float* __restrict__ out) {
  if (threadIdx.x == 0 && blockIdx.x == 0) {
    const float cd  = acc[0] / (float)(BATCH * NPTS);
    const float uni = acc[1] / (float)(BATCH * NPTS * 4);
    out[0] = cd + 0.1f * uni;
  }
}

extern "C" void kernel_launch(void* const* d_in, const int* in_sizes, int n_in,
                              void* d_out, int out_size, void* d_ws, size_t ws_size,
                              hipStream_t stream) {
  (void)in_sizes; (void)n_in; (void)out_size; (void)ws_size;
  const float* pred = (const float*)d_in[0];   // (B, N, 3) f32
  const float* gt   = (const float*)d_in[1];   // (B, N, 3) f32
  const float* rad  = (const float*)d_in[2];   // (B, 1)    f32
  float* out = (float*)d_out;
  float* acc = (float*)d_ws;                   // acc[0]=chamfer sum, acc[1]=uniform sum

  init_kernel<<<1, 32, 0, stream>>>(acc);

  dim3 grid(BATCH * COLTILES);                 // 2048 blocks, 16 rows each
  chamfer_min_kernel<<<grid, 256, 0, stream>>>(gt,   pred, rad, acc, 0.8f);
  chamfer_min_kernel<<<grid, 256, 0, stream>>>(pred, gt,   rad, acc, 0.2f);
  uniform_topk_kernel<<<grid, 256, 0, stream>>>(pred, acc);
  finalize_kernel<<<1, 32, 0, stream>>>(acc, out);
}